// MotionCorrect_36687610642690
// MI455X (gfx1250) — compile-verified
//
#include <hip/hip_runtime.h>
#include <math.h>

// ---------------------------------------------------------------------------
// MotionCorrect for MI455X (gfx1250, wave32).
//
// Heavy op: 1024x1024 (x) 1004x1004 VALID cross-correlation -> 21x21 map.
// Mapped to f32 WMMA (V_WMMA_F32_16X16X4_F32) as a Toeplitz GEMM:
//   num[i,j] = sum_u sum_w Xzm[i+u, w] * Tzm[u, w-j]
// M=i (21->32), N=j (21->32), K=(u,w) ~ 1.03M, accumulated in v8f WMMA accs.
//
// VGPR layouts per CDNA5 ISA 7.12.2 (wave32):
//   A 16x4 f32 : lane m (0..15) & lane m+16 ; VGPR g holds K = g + 2*(lane>>4)
//   C/D 16x16  : VGPR g holds M = g + 8*(lane>>4), N = lane&15
//   B 4x16     : assumed mirror of A (VGPR g: K = g + 2*(lane>>4), N = lane&15)
// ---------------------------------------------------------------------------

typedef float v2f __attribute__((ext_vector_type(2)));
typedef float v8f __attribute__((ext_vector_type(8)));

#define HH 2048
#define CC 512          // C_X = C_Y
#define OFFC 522        // OFF_X = OFF_Y
#define KH 1004         // template crop size (kh = kw)
#define NW 21           // ncc map size
#define MSH 10
#define EPSF 1e-8f

// correlation kernel tiling
#define UPB 16          // template rows per workgroup
#define KB 128          // image cols per K-block
#define TPAD 31
#define TW (KB + 32)    // 160: template-row stage width (31 left pad + 128 + 1)
#define XROWS 47        // UPB-1 + 32 output rows
#define XSTR 130        // padded LDS row stride (floats)
#define NBLK_U ((KH + UPB - 1) / UPB)   // 63

// workspace layout (floats):
//  [0]=sumT [1]=sumT2 [2]=sumX [3]=meanX [4]=meanT [5]=tmpl_var [6]=xs [7]=ys
//  [16..457)    num (441, atomically accumulated)
//  [464..905)   localsum (441)
//  [912..1353)  localsum_sq (441)
//  [2048..)     P1 row window sums (1024*21)
//  [23552..)    P2 row window sums of squares (1024*21)
// total 45056 floats = 176 KB of d_ws.
#define WS_NUM 16
#define WS_LS  464
#define WS_LS2 912
#define WS_P1  2048
#define WS_P2  23552

// ---------------------------------------------------------------------------
__global__ void k_init(float* ws) {
    for (int i = threadIdx.x; i < 2048; i += blockDim.x) ws[i] = 0.f;
}

// sum / sumsq over template[522:1526, 522:1526]
__global__ void k_reduce_tmpl(const float* __restrict__ T, float* ws) {
    __shared__ float s1[8], s2[8];
    float a = 0.f, b = 0.f;
    const int N = KH * KH;
    for (int i = blockIdx.x * blockDim.x + threadIdx.x; i < N;
         i += gridDim.x * blockDim.x) {
        int u = i / KH, v = i - u * KH;
        float t = T[(size_t)(OFFC + u) * HH + OFFC + v];
        a += t; b += t * t;
    }
    for (int o = 16; o > 0; o >>= 1) { a += __shfl_down(a, o, 32); b += __shfl_down(b, o, 32); }
    int w = threadIdx.x >> 5, l = threadIdx.x & 31;
    if (l == 0) { s1[w] = a; s2[w] = b; }
    __syncthreads();
    if (threadIdx.x == 0) {
        float A = 0.f, B = 0.f;
        for (int i = 0; i < (int)(blockDim.x >> 5); ++i) { A += s1[i]; B += s2[i]; }
        atomicAdd(&ws[0], A); atomicAdd(&ws[1], B);
    }
}

// sum over X[0, 512:1536, 512:1536, 0]
__global__ void k_reduce_x(const float* __restrict__ X, float* ws) {
    __shared__ float s1[8];
    float a = 0.f;
    const int N = 1024 * 1024;
    for (int i = blockIdx.x * blockDim.x + threadIdx.x; i < N;
         i += gridDim.x * blockDim.x) {
        int r = i >> 10, c = i & 1023;
        a += X[(size_t)(CC + r) * HH + CC + c];
    }
    for (int o = 16; o > 0; o >>= 1) a += __shfl_down(a, o, 32);
    int w = threadIdx.x >> 5, l = threadIdx.x & 31;
    if (l == 0) s1[w] = a;
    __syncthreads();
    if (threadIdx.x == 0) {
        float A = 0.f;
        for (int i = 0; i < (int)(blockDim.x >> 5); ++i) A += s1[i];
        atomicAdd(&ws[2], A);
    }
}

__global__ void k_finalize(float* ws) {
    const float nT = (float)KH * (float)KH;
    ws[3] = ws[2] / (1024.f * 1024.f);              // meanX
    ws[4] = ws[0] / nT;                             // meanT
    ws[5] = ws[1] - ws[0] * ws[0] / nT + EPSF;      // tmpl_var
}

// per-row sliding window sums: P1[r][j] = sum_{v<1004} Xc[r, j+v], P2 squares
__global__ void k_rowsums(const float* __restrict__ X, float* ws) {
    __shared__ float x[1024];
    __shared__ float s1[8], s2[8];
    const int r = blockIdx.x;                       // 0..1023
    for (int c = threadIdx.x; c < 1024; c += blockDim.x)
        x[c] = X[(size_t)(CC + r) * HH + CC + c];
    __syncthreads();
    float a = 0.f, b = 0.f;
    for (int c = threadIdx.x; c < KH; c += blockDim.x) { float v = x[c]; a += v; b += v * v; }
    for (int o = 16; o > 0; o >>= 1) { a += __shfl_down(a, o, 32); b += __shfl_down(b, o, 32); }
    int w = threadIdx.x >> 5, l = threadIdx.x & 31;
    if (l == 0) { s1[w] = a; s2[w] = b; }
    __syncthreads();
    if (threadIdx.x == 0) {
        float S1 = 0.f, S2 = 0.f;
        for (int i = 0; i < (int)(blockDim.x >> 5); ++i) { S1 += s1[i]; S2 += s2[i]; }
        float* P1 = ws + WS_P1; float* P2 = ws + WS_P2;
        P1[r * NW] = S1; P2[r * NW] = S2;
        for (int j = 1; j < NW; ++j) {
            float ad = x[j + KH - 1], sb = x[j - 1];
            S1 += ad - sb; S2 += ad * ad - sb * sb;
            P1[r * NW + j] = S1; P2[r * NW + j] = S2;
        }
    }
}

// ls[i,j] = sum_u P1[i+u][j] ; ls2 likewise
__global__ void k_colsums(float* ws) {
    __shared__ float s1[4], s2[4];
    const float* P1 = ws + WS_P1; const float* P2 = ws + WS_P2;
    const int o = blockIdx.x, i = o / NW, j = o - i * NW;
    float a = 0.f, b = 0.f;
    for (int u = threadIdx.x; u < KH; u += blockDim.x) {
        a += P1[(i + u) * NW + j]; b += P2[(i + u) * NW + j];
    }
    for (int off = 16; off > 0; off >>= 1) { a += __shfl_down(a, off, 32); b += __shfl_down(b, off, 32); }
    int w = threadIdx.x >> 5, l = threadIdx.x & 31;
    if (l == 0) { s1[w] = a; s2[w] = b; }
    __syncthreads();
    if (threadIdx.x == 0) {
        float A = 0.f, B = 0.f;
        for (int k = 0; k < (int)(blockDim.x >> 5); ++k) { A += s1[k]; B += s2[k]; }
        ws[WS_LS + o] = A; ws[WS_LS2 + o] = B;
    }
}

// ---------------------------------------------------------------------------
// WMMA correlation: each workgroup owns 16 template rows (4 waves x 4 rows),
// accumulates a 32x32 f32 tile in WMMA accumulators over K = 16 rows x 1024.
__global__ void __launch_bounds__(128)
k_corr_wmma(const float* __restrict__ X, const float* __restrict__ T,
            float* __restrict__ ws) {
    __shared__ float sX[XROWS * XSTR];   // image slab (Xc - meanX)
    __shared__ float sT[UPB * TW];       // zero-padded template rows (Tzm)

    const float meanX = ws[3];
    const float meanT = ws[4];
    float* num = ws + WS_NUM;

    const int u0   = blockIdx.x * UPB;
    const int tid  = threadIdx.x;
    const int wave = tid >> 5;
    const int lane = tid & 31;
    const int n    = lane & 15;
    const int hi   = lane >> 4;

    v8f acc00 = {}, acc01 = {}, acc10 = {}, acc11 = {};

    for (int kb = 0; kb < 1024 / KB; ++kb) {
        const int kg0 = kb * KB;
        __syncthreads();
        // stage image slab: rows u0..u0+46 of Xc, cols kg0..kg0+127 (coalesced)
        for (int r = 0; r < XROWS; ++r) {
            int gr = u0 + r;   // max 992+46=1038 -> X row 1550, in bounds
            sX[r * XSTR + tid] =
                X[(size_t)(CC + gr) * HH + (CC + kg0 + tid)] - meanX;
        }
        // stage template rows with 31-col left pad; zero outside window / u>=KH
        for (int idx = tid; idx < UPB * TW; idx += 128) {
            int uu = idx / TW, s = idx - uu * TW;
            int u = u0 + uu;
            int w = kg0 - TPAD + s;
            float v = 0.f;
            if (u < KH && w >= 0 && w < KH)
                v = T[(size_t)(OFFC + u) * HH + (OFFC + w)] - meanT;
            sT[uu * TW + s] = v;
        }
        __syncthreads();

        for (int uu = 0; uu < 4; ++uu) {
            const int uIdx = wave * 4 + uu;           // 0..15
            const float* tRow = sT + uIdx * TW;
            for (int kl = 0; kl < KB; kl += 4) {
                // A tiles: A[M,K] = Xzm[u+M, kg0+kl+K], K = g + 2*hi
                const int acol = kl + 2 * hi;
                const float* ar0 = sX + (n + uIdx) * XSTR + acol;       // M=n
                const float* ar1 = sX + (n + 16 + uIdx) * XSTR + acol;  // M=n+16
                v2f a0 = { ar0[0], ar0[1] };
                v2f a1 = { ar1[0], ar1[1] };
                // B tiles: B[K,N] = Tzm[u, (kl+K) - j], j = n (+16 for tj=1)
                const int sb = kl + 2 * hi - n + TPAD;  // in [16,158]
                v2f b0 = { tRow[sb],      tRow[sb + 1]      };
                v2f b1 = { tRow[sb - 16], tRow[sb - 15]     };
                acc00 = __builtin_amdgcn_wmma_f32_16x16x4_f32(
                    false, a0, false, b0, (short)0, acc00, false, false);
                acc01 = __builtin_amdgcn_wmma_f32_16x16x4_f32(
                    false, a0, false, b1, (short)0, acc01, false, false);
                acc10 = __builtin_amdgcn_wmma_f32_16x16x4_f32(
                    false, a1, false, b0, (short)0, acc10, false, false);
                acc11 = __builtin_amdgcn_wmma_f32_16x16x4_f32(
                    false, a1, false, b1, (short)0, acc11, false, false);
            }
        }
    }

    // C/D layout: VGPR g -> M = g + 8*hi, N = lane&15 ; tiles add 16.
    for (int g = 0; g < 8; ++g) {
        int M0 = g + 8 * hi;
        int M1 = M0 + 16;
        int N0 = n, N1 = n + 16;
        if (M0 < NW) {
            if (N0 < NW) atomicAdd(&num[M0 * NW + N0], acc00[g]);
            if (N1 < NW) atomicAdd(&num[M0 * NW + N1], acc01[g]);
        }
        if (M1 < NW) {
            if (N0 < NW) atomicAdd(&num[M1 * NW + N0], acc10[g]);
            if (N1 < NW) atomicAdd(&num[M1 * NW + N1], acc11[g]);
        }
    }
}

// ---------------------------------------------------------------------------
// NCC map -> argmax -> subpixel quadratic (log) fit. Single block.
__global__ void k_ncc_peak(float* ws) {
    __shared__ float ncc[NW * NW];
    __shared__ float rv[512];
    __shared__ int   ri[512];
    const int tid = threadIdx.x;
    const float tvar = ws[5];
    float v = -3.4e38f;
    if (tid < NW * NW) {
        float nm  = ws[WS_NUM + tid];
        float ls  = ws[WS_LS + tid];
        float ls2 = ws[WS_LS2 + tid];
        float iv = ls2 - ls * ls / ((float)KH * (float)KH) + EPSF;
        if (iv < 0.f) iv = 0.f;
        float c = nm / sqrtf(tvar * iv);
        if (__builtin_isnan(c)) c = 0.f;
        ncc[tid] = c;
        v = c;
    }
    rv[tid] = v; ri[tid] = tid;
    __syncthreads();
    for (int s = 256; s > 0; s >>= 1) {
        if (tid < s) {
            if (rv[tid + s] > rv[tid] ||
                (rv[tid + s] == rv[tid] && ri[tid + s] < ri[tid])) {
                rv[tid] = rv[tid + s]; ri[tid] = ri[tid + s];
            }
        }
        __syncthreads();
    }
    if (tid == 0) {
        int ami = ri[0];
        int sx = ami / NW, sy = ami - sx * NW;
        auto m21 = [](int a) { return ((a % NW) + NW) % NW; };
        float l0  = logf(ncc[sx * NW + sy]);
        float lxm = logf(ncc[m21(sx - 1) * NW + sy]);
        float lxp = logf(ncc[m21(sx + 1) * NW + sy]);
        float lym = logf(ncc[sx * NW + m21(sy - 1)]);
        float lyp = logf(ncc[sx * NW + m21(sy + 1)]);
        float shxn = -(float)(sx - MSH);
        float shyn = -(float)(sy - MSH);
        ws[6] = shxn - (lxm - lxp) / (2.f * lxm - 4.f * l0 + 2.f * lxp);
        ws[7] = shyn - (lym - lyp) / (2.f * lym - 4.f * l0 + 2.f * lyp);
    }
}

// bilinear resample of the full 2048x2048 image by (xs, ys)
__global__ void k_resample(const float* __restrict__ X, const float* __restrict__ ws,
                           float* __restrict__ out) {
    const int idx = blockIdx.x * blockDim.x + threadIdx.x;
    if (idx >= HH * HH) return;
    const int r = idx >> 11, c = idx & (HH - 1);
    const float xs = ws[6], ys = ws[7];
    float rr = (float)r - xs, cc = (float)c - ys;
    float r0f = floorf(rr), c0f = floorf(cc);
    int r0 = (int)r0f, c0 = (int)c0f;
    float fr = rr - r0f, fc = cc - c0f;
    auto samp = [&](int ri, int ci) -> float {
        if (ri < 0 || ri >= HH || ci < 0 || ci >= HH) return 0.f;
        return X[(size_t)ri * HH + ci];
    };
    out[idx] = (1.f - fr) * (1.f - fc) * samp(r0, c0)
             + (1.f - fr) * fc         * samp(r0, c0 + 1)
             + fr         * (1.f - fc) * samp(r0 + 1, c0)
             + fr         * fc         * samp(r0 + 1, c0 + 1);
}

// ---------------------------------------------------------------------------
extern "C" void kernel_launch(void* const* d_in, const int* in_sizes, int n_in,
                              void* d_out, int out_size, void* d_ws, size_t ws_size,
                              hipStream_t stream) {
    const float* X = (const float*)d_in[0];   // (1,2048,2048,1) f32
    const float* T = (const float*)d_in[1];   // (2048,2048) f32
    float* out = (float*)d_out;               // (1, 2048*2048) f32
    float* ws  = (float*)d_ws;                // >= 176 KB used

    k_init       <<<1, 512, 0, stream>>>(ws);
    k_reduce_tmpl<<<1024, 256, 0, stream>>>(T, ws);
    k_reduce_x   <<<1024, 256, 0, stream>>>(X, ws);
    k_finalize   <<<1, 1, 0, stream>>>(ws);
    k_rowsums    <<<1024, 256, 0, stream>>>(X, ws);
    k_colsums    <<<NW * NW, 128, 0, stream>>>(ws);
    k_corr_wmma  <<<NBLK_U, 128, 0, stream>>>(X, T, ws);
    k_ncc_peak   <<<1, 512, 0, stream>>>(ws);
    k_resample   <<<(HH * HH + 255) / 256, 256, 0, stream>>>(X, ws, out);
}